// SoftDecisionTree_86440511799669
// MI455X (gfx1250) — compile-verified
//
#include <hip/hip_runtime.h>

typedef __attribute__((ext_vector_type(2))) float v2f;
typedef __attribute__((ext_vector_type(8))) float v8f;

#define BATCH   16384
#define NFEAT   1024
#define TDEPTH  10
#define NLEAVES 1024
#define WPB     8                 // waves per block
#define RPW     16                // rows per wave
#define RPB     (WPB * RPW)       // 128 rows per block
#define KCHUNKS (NFEAT / 4)       // 256 wmma k-steps

__launch_bounds__(256)
__global__ void sdt_wmma_kernel(const float* __restrict__ x,
                                const float* __restrict__ W,
                                const float* __restrict__ bias,
                                const float* __restrict__ leaf,
                                float* __restrict__ out) {
  // B fragments: chunk c, lane lb, pair j -> W[(lb&15)][4c + 2*(lb>>4) + j]
  __shared__ float ldsB[KCHUNKS * 64];   // 64 KB
  __shared__ float ldsLeaf[NLEAVES];     // 4 KB
  __shared__ float ldsZ[WPB * 256];      // 8 KB  (z tile transpose)
  __shared__ float ldsV[WPB * 32];       // 1 KB  (half-fold values)
  __shared__ float ldsBias[16];

  const int t    = threadIdx.x;
  const int wave = t >> 5;
  const int lane = t & 31;

  // ---- stage W (first 16 rows) as pre-swizzled B fragments, leaves, bias ----
  for (int idx = t; idx < 16 * NFEAT; idx += 256) {
    int n = idx >> 10;
    int k = idx & (NFEAT - 1);
    float w = W[n * NFEAT + k];
    int c  = k >> 2;
    int lb = ((k >> 1) & 1) * 16 + n;
    int j  = k & 1;
    ldsB[c * 64 + lb * 2 + j] = w;
  }
  for (int idx = t; idx < NLEAVES; idx += 256) ldsLeaf[idx] = leaf[idx];
  if (t < 16) ldsBias[t] = bias[t];
  __syncthreads();

  const int rowBase = (blockIdx.x * WPB + wave) * RPW;
  const int m = lane & 15;     // row within tile (A fragment) / later: my row
  const int h = lane >> 4;     // K-pair half for A/B; later: leaf-tree half

  // ---- GEMM: z[16x16] = x_tile[16x1024] * W16^T via V_WMMA_F32_16X16X4_F32 ----
  const float* xrow = x + (size_t)(rowBase + m) * NFEAT + h * 2;
  const float* bptr = ldsB + lane * 2;
  v8f acc = {};
  #pragma unroll 4
  for (int c = 0; c < KCHUNKS; ++c) {
    v2f a  = *(const v2f*)(xrow + c * 4);
    v2f bf = *(const v2f*)(bptr + c * 64);
    acc = __builtin_amdgcn_wmma_f32_16x16x4_f32(
        /*neg_a=*/false, a, /*neg_b=*/false, bf,
        /*c_mod=*/(short)0, acc, /*reuse_a=*/false, /*reuse_b=*/false);
  }

  // ---- transpose C through LDS: lane holds C[M=i+8h][N=lane&15] in acc[i] ----
  {
    float* zp = ldsZ + wave * 256;
    int n  = lane & 15;
    int mh = h * 8;
    #pragma unroll
    for (int i = 0; i < 8; ++i) zp[(mh + i) * 16 + n] = acc[i];
  }
  __syncthreads();

  // ---- gates for my row ----
  float g[TDEPTH], gm[TDEPTH];
  {
    const float* zp = ldsZ + wave * 256 + m * 16;
    #pragma unroll
    for (int d = 0; d < TDEPTH; ++d) {
      float z = zp[d] + ldsBias[d];
      float e = __expf(-z);
      g[d]  = 1.0f / (1.0f + e);
      gm[d] = 1.0f - g[d];
    }
  }

  // ---- fold 512 leaves (half h of the tree) with g9..g1, fully in registers ----
  const float* lf = ldsLeaf + h * 512;
  float c16[16];
  #pragma unroll
  for (int jj = 0; jj < 16; ++jj) {          // 16 chunks of 32 leaves
    float tt[16];
    #pragma unroll
    for (int i = 0; i < 16; ++i) {
      v2f lv = *(const v2f*)(lf + jj * 32 + i * 2);
      tt[i] = gm[9] * lv.x + g[9] * lv.y;    // level 9 (LSB of leaf id)
    }
    #pragma unroll
    for (int i = 0; i < 8; ++i) tt[i] = gm[8] * tt[2*i] + g[8] * tt[2*i+1];
    #pragma unroll
    for (int i = 0; i < 4; ++i) tt[i] = gm[7] * tt[2*i] + g[7] * tt[2*i+1];
    #pragma unroll
    for (int i = 0; i < 2; ++i) tt[i] = gm[6] * tt[2*i] + g[6] * tt[2*i+1];
    c16[jj] = gm[5] * tt[0] + g[5] * tt[1];
  }
  #pragma unroll
  for (int i = 0; i < 8; ++i) c16[i] = gm[4] * c16[2*i] + g[4] * c16[2*i+1];
  #pragma unroll
  for (int i = 0; i < 4; ++i) c16[i] = gm[3] * c16[2*i] + g[3] * c16[2*i+1];
  #pragma unroll
  for (int i = 0; i < 2; ++i) c16[i] = gm[2] * c16[2*i] + g[2] * c16[2*i+1];
  float vh = gm[1] * c16[0] + g[1] * c16[1];

  // ---- combine the two halves of each row with g0, write output ----
  ldsV[wave * 32 + h * 16 + m] = vh;
  __syncthreads();
  if (lane < 16) {
    float v0 = ldsV[wave * 32 + m];        // MSB bit = 0 half
    float v1 = ldsV[wave * 32 + 16 + m];   // MSB bit = 1 half
    out[rowBase + m] = gm[0] * v0 + g[0] * v1;
  }
}

extern "C" void kernel_launch(void* const* d_in, const int* in_sizes, int n_in,
                              void* d_out, int out_size, void* d_ws, size_t ws_size,
                              hipStream_t stream) {
  (void)in_sizes; (void)n_in; (void)out_size; (void)d_ws; (void)ws_size;
  const float* x    = (const float*)d_in[0];
  const float* W    = (const float*)d_in[1];
  const float* b    = (const float*)d_in[2];
  const float* leaf = (const float*)d_in[3];
  float* out = (float*)d_out;
  sdt_wmma_kernel<<<dim3(BATCH / RPB), dim3(256), 0, stream>>>(x, W, b, leaf, out);
}